// GraphAttention_18167711662488
// MI455X (gfx1250) — compile-verified
//
#include <hip/hip_runtime.h>
#include <hip/hip_bf16.h>

// GAT edge-softmax for MI455X (gfx1250, wave32).
// HIDDEN=64, NHEADS=8, HEAD_DIM=8. Memory-bound: k/q/v each read exactly once.
// Pass 1 computes per-edge head scores with V_WMMA_F32_16X16X4_F32:
//   scores(16x16) = P(16x64) x S(64x16), P = k (elementwise*) q, S = head indicator.

typedef float v2f __attribute__((ext_vector_type(2)));
typedef float v8f __attribute__((ext_vector_type(8)));

#define HIDDEN 64
#define NHEADS 8
#define ROWP 68  // LDS row pitch in floats: banks (4m+kk)%64 -> conflict-free, 16B aligned

__device__ __forceinline__ unsigned float_key(float f) {
  unsigned u = __float_as_uint(f);
  return (u & 0x80000000u) ? ~u : (u | 0x80000000u);  // monotonic uint ordering
}
__device__ __forceinline__ float key_float(unsigned k) {
  return (k & 0x80000000u) ? __uint_as_float(k ^ 0x80000000u) : __uint_as_float(~k);
}

// ---- Pass 1: scores (WMMA) + segment-max via uint atomicMax -----------------
__global__ __launch_bounds__(256) void gat_scores_wmma(
    const float* __restrict__ keys, const float* __restrict__ queries,
    const int* __restrict__ dst, float* __restrict__ scores,
    unsigned* __restrict__ maxkey, int E) {
  __shared__ float lds[8 * 16 * ROWP];  // 8 waves x (16 edges x 68 floats)
  const int lane = threadIdx.x & 31;
  const int wave = threadIdx.x >> 5;
  float* p = lds + wave * (16 * ROWP);
  const int e0 = (blockIdx.x * 8 + wave) * 16;

  // Coalesced load of a 16x64 k/q tile; store p = k*q into padded LDS rows.
  #pragma unroll
  for (int i = 0; i < 8; ++i) {
    int f = i * 128 + lane * 4;          // flat float4 index within tile
    int row = f >> 6, col = f & 63;
    int e = e0 + row; if (e >= E) e = E - 1;          // branchless clamp: EXEC stays full
    const float4 k4 = *(const float4*)(keys + (size_t)e * HIDDEN + col);
    const float4 q4 = *(const float4*)(queries + (size_t)e * HIDDEN + col);
    float4 p4 = make_float4(k4.x * q4.x, k4.y * q4.y, k4.z * q4.z, k4.w * q4.w);
    *(float4*)(p + row * ROWP + col) = p4;
  }
  __syncthreads();

  // scores(16x16) = sum_kk A(16x4) * B(4x16);  B chunk = head-indicator column.
  const int m = lane & 15;           // A-matrix M row = edge-in-tile
  const int half2 = (lane >> 4) << 1;  // K slots {0,1} vs {2,3} per ISA A layout
  v8f c = {};
  #pragma unroll
  for (int kk = 0; kk < HIDDEN; kk += 4) {
    const float bv = (m == (kk >> 3)) ? 1.0f : 0.0f;  // all 4 B rows identical
    v2f a = *(const v2f*)(p + m * ROWP + kk + half2); // dims kk+half2, kk+half2+1
    v2f b; b.x = bv; b.y = bv;
    c = __builtin_amdgcn_wmma_f32_16x16x4_f32(false, a, false, b, (short)0, c,
                                              false, false);
  }

  // C layout: VGPR r -> lanes 0-15: (M=r, N=lane), lanes 16-31: (M=r+8, N=lane-16).
  // Valid head columns are N=0..7.
  const bool active = (lane & 8) == 0;
  const int h = lane & 7;
  #pragma unroll
  for (int r = 0; r < 8; ++r) {
    float s = c[r] * 0.125f;                  // HIDDEN^-0.5
    s = (s >= 0.0f) ? s : 0.2f * s;           // LeakyReLU(0.2)
    int e = e0 + r + ((lane >> 4) << 3);
    if (active && e < E) {
      scores[(size_t)e * NHEADS + h] = s;
      atomicMax(&maxkey[(size_t)dst[e] * NHEADS + h], float_key(s));
    }
  }
}

// ---- Pass 2: ex = exp(s - max[dst]); denom[dst] += ex -----------------------
__global__ void gat_exp(const int* __restrict__ dst, float* __restrict__ scores,
                        float* __restrict__ denom,
                        const unsigned* __restrict__ maxkey, int EH) {
  int t = blockIdx.x * blockDim.x + threadIdx.x;
  if (t >= EH) return;
  int e = t >> 3, h = t & 7;
  int d = dst[e];
  float mx = key_float(maxkey[(size_t)d * NHEADS + h]);
  float ex = __expf(scores[t] - mx);
  scores[t] = ex;                                    // overwrite in place
  atomicAdd(&denom[(size_t)d * NHEADS + h], ex);
}

// ---- Pass 3: out[dst] += (ex/denom[dst]) * v --------------------------------
__global__ void gat_scatter(const int* __restrict__ dst,
                            const float* __restrict__ ex,
                            const float* __restrict__ denom,
                            const float* __restrict__ values,
                            float* __restrict__ out, long ED) {
  long t = (long)blockIdx.x * blockDim.x + threadIdx.x;
  if (t >= ED) return;
  long e = t >> 6; int dim = (int)(t & 63); int h = dim >> 3;
  int d = dst[e];
  float w = ex[e * NHEADS + h] *
            __builtin_amdgcn_rcpf(denom[(size_t)d * NHEADS + h]);
  atomicAdd(&out[(size_t)d * HIDDEN + dim], w * values[t]);
}

extern "C" void kernel_launch(void* const* d_in, const int* in_sizes, int n_in,
                              void* d_out, int out_size, void* d_ws, size_t ws_size,
                              hipStream_t stream) {
  const float* keys    = (const float*)d_in[0];
  const float* queries = (const float*)d_in[1];
  const float* values  = (const float*)d_in[2];
  const int*   dst     = (const int*)d_in[3];
  const long E = in_sizes[0] / HIDDEN;      // 800000
  const long N = out_size / HIDDEN;         // 50000

  // ws layout: [E*8 f32 scores/ex][N*8 u32 maxkey][N*8 f32 denom] ~ 28.8 MB
  float*    scores = (float*)d_ws;
  unsigned* maxkey = (unsigned*)((char*)d_ws + (size_t)E * NHEADS * 4);
  float*    denom  = (float*)((char*)d_ws + (size_t)(E + N) * NHEADS * 4);
  float*    out    = (float*)d_out;

  hipMemsetAsync(out, 0, (size_t)out_size * sizeof(float), stream);
  hipMemsetAsync(maxkey, 0, (size_t)N * NHEADS * 2 * 4, stream);  // maxkey + denom

  const long tiles = (E + 15) / 16;
  const int blocks1 = (int)((tiles + 7) / 8);
  gat_scores_wmma<<<blocks1, 256, 0, stream>>>(keys, queries, dst, scores,
                                               maxkey, (int)E);

  const long EH = E * NHEADS;
  gat_exp<<<(int)((EH + 255) / 256), 256, 0, stream>>>(dst, scores, denom,
                                                       maxkey, (int)EH);

  const long ED = E * HIDDEN;
  gat_scatter<<<(int)((ED + 255) / 256), 256, 0, stream>>>(dst, scores, denom,
                                                           values, out, ED);
}